// MiniBatchDiscrimination_63771674411203
// MI455X (gfx1250) — compile-verified
//
#include <hip/hip_runtime.h>

typedef __attribute__((ext_vector_type(2))) float v2f;
typedef __attribute__((ext_vector_type(8))) float v8f;

#define B_SZ   512
#define D_SZ   1024
#define NKER   160
#define NCOLS  480            // 160*3
#define OUTC   1184           // 1024 + 160
#define TILE_J 32             // 32*480*4 = 60 KB LDS

// ---------------------------------------------------------------------------
// Kernel 1: act = inputs @ W + b  via V_WMMA_F32_16X16X4_F32
// One wave32 per 16x16 output tile; K-loop over 1024 in steps of 4.
// A-frag layout (ISA 7.12.2, 32-bit A 16x4): lane<16 -> K={0,1}, lane>=16 -> K={2,3}
// C/D layout: VGPR r: lanes0-15 M=r, lanes16-31 M=8+r, N=lane%16
// ---------------------------------------------------------------------------
__global__ __launch_bounds__(32)
void gemm_wmma_f32(const float* __restrict__ A, const float* __restrict__ W,
                   const float* __restrict__ bias, float* __restrict__ act)
{
    const int m0   = blockIdx.x * 16;
    const int n0   = blockIdx.y * 16;
    const int lane = threadIdx.x;        // 0..31, full wave -> EXEC all ones
    const int half = lane >> 4;          // 0: K lo pair, 1: K hi pair
    const int l16  = lane & 15;

    const float* aptr = A + (m0 + l16) * D_SZ + 2 * half;   // contiguous K pair
    const float* bptr = W + (2 * half) * NCOLS + (n0 + l16);

    v8f c = {};
#pragma unroll 8
    for (int kk = 0; kk < D_SZ; kk += 4) {
        v2f a = *(const v2f*)(aptr + kk);          // A[m][kk+2h .. kk+2h+1]
        v2f b;
        b.x = bptr[(kk    ) * NCOLS];              // W[kk+2h  ][n]
        b.y = bptr[(kk + 1) * NCOLS];              // W[kk+2h+1][n]
        // 8 args: (neg_a, A, neg_b, B, c_mod, C, reuse_a, reuse_b)
        c = __builtin_amdgcn_wmma_f32_16x16x4_f32(false, a, false, b,
                                                  (short)0, c, false, false);
    }

    const int ncol = n0 + l16;
    const float bv = bias[ncol];
#pragma unroll
    for (int r = 0; r < 8; ++r) {
        const int m = m0 + r + 8 * half;
        act[m * NCOLS + ncol] = c[r] + bv;
    }
}

// ---------------------------------------------------------------------------
// Kernel 2: copy inputs into out[:, 0:1024] (row stride 1184), float4 moves
// ---------------------------------------------------------------------------
__global__ __launch_bounds__(256)
void copy_inputs(const float4* __restrict__ in, float* __restrict__ out)
{
    const int i = blockIdx.x;
    const int t = threadIdx.x;                 // 256 threads * float4 = 1024 cols
    float4 v = in[i * (D_SZ / 4) + t];
    *(float4*)(out + i * OUTC + t * 4) = v;    // 1184 % 4 == 0 -> 16B aligned
}

// ---------------------------------------------------------------------------
// Kernel 3: features[i,k] = sum_j exp(-L1(act[i,k,:], act[j,k,:]))
// One block per row i; 160 threads (5 waves), thread k owns kernel k.
// act[j] rows staged through LDS in 32-row tiles (contiguous 60 KB chunks).
// ---------------------------------------------------------------------------
__global__ __launch_bounds__(NKER)
void features_kernel(const float* __restrict__ act, float* __restrict__ out)
{
    __shared__ float lds[TILE_J * NCOLS];

    const int i = blockIdx.x;
    const int k = threadIdx.x;                 // 0..159

    const float a0 = act[i * NCOLS + 3 * k + 0];
    const float a1 = act[i * NCOLS + 3 * k + 1];
    const float a2 = act[i * NCOLS + 3 * k + 2];

    float acc = 0.0f;
    for (int j0 = 0; j0 < B_SZ; j0 += TILE_J) {
        __syncthreads();
        // full rows -> one contiguous 15360-float chunk, coalesced across threads
        const float* src = act + j0 * NCOLS;
        for (int idx = k; idx < TILE_J * NCOLS; idx += NKER)
            lds[idx] = src[idx];
        __syncthreads();

        if (j0 + TILE_J < B_SZ)                               // warm L2/WGP$ for next tile
            __builtin_prefetch(act + (j0 + TILE_J) * NCOLS + k, 0, 1);

#pragma unroll 4
        for (int j = 0; j < TILE_J; ++j) {
            const float* r = lds + j * NCOLS + 3 * k;         // stride-3: conflict-free
            float d = fabsf(a0 - r[0]) + fabsf(a1 - r[1]) + fabsf(a2 - r[2]);
            acc += __expf(-d);                                // v_exp_f32
        }
    }
    out[i * OUTC + D_SZ + k] = acc;
}

// ---------------------------------------------------------------------------
extern "C" void kernel_launch(void* const* d_in, const int* in_sizes, int n_in,
                              void* d_out, int out_size, void* d_ws, size_t ws_size,
                              hipStream_t stream)
{
    const float* inputs = (const float*)d_in[0];   // [512,1024] f32
    const float* W      = (const float*)d_in[1];   // [1024,480] f32
    const float* bias   = (const float*)d_in[2];   // [480] f32
    float*       out    = (float*)d_out;           // [512,1184] f32
    float*       act    = (float*)d_ws;            // [512,480] f32 scratch (983 KB)

    if (ws_size < (size_t)B_SZ * NCOLS * sizeof(float)) return;

    dim3 gGemm(B_SZ / 16, NCOLS / 16);             // 32 x 30 tiles
    gemm_wmma_f32<<<gGemm, 32, 0, stream>>>(inputs, W, bias, act);

    copy_inputs<<<B_SZ, 256, 0, stream>>>((const float4*)inputs, out);

    features_kernel<<<B_SZ, NKER, 0, stream>>>(act, out);
}